// FeatureSeparationLoss_42838003810561
// MI455X (gfx1250) — compile-verified
//
#include <hip/hip_runtime.h>

typedef __attribute__((ext_vector_type(2))) float v2f;
typedef __attribute__((ext_vector_type(4))) float v4f;
typedef __attribute__((ext_vector_type(8))) float v8f;

// Problem dims (fixed by the reference): B=4, C=256, H=W=256.
#define NCHAN    1024      // B*C
#define CH_ELEMS 65536     // H*W, contiguous per channel in (B,C,H,W) layout
#define BLK      256       // 8 waves of 32

// Kernel 1: per-channel sum over H*W. Bandwidth-bound: 268 MB streamed once
// (bigger than the 192 MB L2 -> stream with non-temporal loads).
// Each block owns one channel (256 KB, contiguous). 64 x 128-bit NT loads per
// thread; cross-lane reduction done on the matrix engine via
// V_WMMA_F32_16X16X4_F32 with an all-ones B matrix (D[m][n] = sum_k A[m][k]).
__global__ __launch_bounds__(BLK)
void channel_sum_kernel(const float* __restrict__ content, float* __restrict__ cs) {
    const int ch = blockIdx.x;           // b*C + c
    const int t  = threadIdx.x;          // 0..255
    const v4f* p =
        reinterpret_cast<const v4f*>(content + (size_t)ch * CH_ELEMS);

    // 65536 floats / 256 threads / 4 per load = 64 iterations, stride-256 v4fs.
    float s0 = 0.0f, s1 = 0.0f;
    #pragma unroll 8
    for (int i = 0; i < 64; ++i) {
        v4f v = __builtin_nontemporal_load(&p[i * BLK + t]);
        s0 += v.x + v.y;
        s1 += v.z + v.w;
    }

    // Matrix-engine lane reduction:
    //   A(16x4): lane L<16  holds (M=L,    K=0,1) = (s0,s1)
    //            lane L>=16 holds (M=L-16, K=2,3) = (s0,s1)
    //   B(4x16) = all ones  =>  D[m][n] = s0_m + s1_m + s0_{m+16} + s1_{m+16}
    v2f a;     a[0] = s0;      a[1] = s1;
    v2f bOne;  bOne[0] = 1.0f; bOne[1] = 1.0f;
    v8f c = {};
    c = __builtin_amdgcn_wmma_f32_16x16x4_f32(
            /*neg_a=*/false, a, /*neg_b=*/false, bOne,
            /*c_mod=*/(short)0, c, /*reuse_a=*/false, /*reuse_b=*/false);

    // Sum the 8 D rows held in this lane, then combine the two half-wave
    // row groups (M=0..7 on lanes 0-15, M=8..15 on lanes 16-31).
    float lane = c[0] + c[1] + c[2] + c[3] + c[4] + c[5] + c[6] + c[7];
    float waveTot = lane + __shfl_xor(lane, 16);

    __shared__ float lds[BLK / 32];
    const int wave = t >> 5;
    if ((t & 31) == 0) lds[wave] = waveTot;
    __syncthreads();
    if (t == 0) {
        float tot = 0.0f;
        #pragma unroll
        for (int w = 0; w < BLK / 32; ++w) tot += lds[w];
        cs[ch] = tot;
    }
}

// Kernel 2: loss = (1/(B*C^2)) * sum_b (sum_i cs[b,i]^2) * (sum_j style[b,j]^2)
// Touches 2048 floats; one wave is plenty.
__global__ __launch_bounds__(32)
void finalize_kernel(const float* __restrict__ cs,
                     const float* __restrict__ style,
                     float* __restrict__ out) {
    const int l = threadIdx.x;  // 0..31
    float x[4] = {0.f, 0.f, 0.f, 0.f};
    float y[4] = {0.f, 0.f, 0.f, 0.f};
    #pragma unroll
    for (int b = 0; b < 4; ++b) {
        for (int i = l; i < 256; i += 32) {
            float cv = cs[b * 256 + i];
            float sv = style[b * 256 + i];
            x[b] += cv * cv;
            y[b] += sv * sv;
        }
    }
    #pragma unroll
    for (int off = 16; off > 0; off >>= 1) {
        #pragma unroll
        for (int b = 0; b < 4; ++b) {
            x[b] += __shfl_xor(x[b], off);
            y[b] += __shfl_xor(y[b], off);
        }
    }
    if (l == 0) {
        float acc = 0.0f;
        #pragma unroll
        for (int b = 0; b < 4; ++b) acc += x[b] * y[b];
        out[0] = acc / (4.0f * 256.0f * 256.0f);
    }
}

extern "C" void kernel_launch(void* const* d_in, const int* in_sizes, int n_in,
                              void* d_out, int out_size, void* d_ws, size_t ws_size,
                              hipStream_t stream) {
    const float* content = (const float*)d_in[0];  // (4,256,256,256) fp32
    const float* style   = (const float*)d_in[1];  // (4,256,1,1)     fp32
    float* cs  = (float*)d_ws;                     // 1024 floats of scratch
    float* out = (float*)d_out;                    // scalar fp32 loss

    channel_sum_kernel<<<NCHAN, BLK, 0, stream>>>(content, cs);
    finalize_kernel<<<1, 32, 0, stream>>>(cs, style, out);
}